// GraphEncoder_32040456029042
// MI455X (gfx1250) — compile-verified
//
#include <hip/hip_runtime.h>
#include <stdint.h>

#define BATCH 64

// ---------------- gfx1250 async-to-LDS helpers ----------------
// GVS mode: global addr = SGPR64 base + VGPR32 byte offset; VDST VGPR holds
// the LDS byte address (relative to LDS_BASE). Tracked with ASYNCcnt.
__device__ __forceinline__ uint32_t lds_off_u32(const void* p) {
    // low 32 bits of a generic pointer to LDS == LDS byte offset
    return (uint32_t)(uintptr_t)p;
}

__device__ __forceinline__ void async_ld16(uint32_t lds_off, const void* sbase, uint32_t voff) {
    asm volatile("global_load_async_to_lds_b128 %0, %1, %2"
                 :: "v"(lds_off), "v"(voff), "s"(sbase)
                 : "memory");
}

__device__ __forceinline__ void wait_async_le1() {
    asm volatile("s_wait_asynccnt 0x1" ::: "memory");
}
__device__ __forceinline__ void wait_async_le0() {
    asm volatile("s_wait_asynccnt 0x0" ::: "memory");
}

// ---------------- zero the output ----------------
__global__ void zero4_kernel(float4* __restrict__ p, size_t n4) {
    size_t i  = (size_t)blockIdx.x * blockDim.x + threadIdx.x;
    size_t st = (size_t)gridDim.x * blockDim.x;
    float4 z; z.x = 0.f; z.y = 0.f; z.z = 0.f; z.w = 0.f;
    for (; i < n4; i += st) p[i] = z;
}

// ---------------- transpose x [64, N] -> xT [N, 64] ----------------
#define TN 32
__global__ __launch_bounds__(256)
void transpose_x_kernel(const float* __restrict__ x, float* __restrict__ xT, int N) {
    __shared__ float tile[BATCH][TN + 1];   // +1 pad: stride 33 -> conflict-free
    const int n0  = blockIdx.x * TN;
    const int tid = threadIdx.x;

    // load: 64(b) x 32(n) tile, coalesced along n
    const int tx = tid & (TN - 1);   // n within tile
    const int t8 = tid >> 5;         // 0..7
    for (int bb = t8; bb < BATCH; bb += 8) {
        const int n = n0 + tx;
        tile[bb][tx] = (n < N) ? x[(size_t)bb * N + n] : 0.f;
    }
    __syncthreads();

    // store: 64 consecutive b per n, coalesced along b
    const int b = tid & (BATCH - 1);
    const int g = tid >> 6;          // 0..3
    for (int i = g; i < TN; i += 4) {
        const int n = n0 + i;
        if (n < N) xT[(size_t)n * BATCH + b] = tile[b][i];
    }
}

// ---------------- edge scatter kernel ----------------
constexpr int CHUNK            = 128;                      // edges per staged chunk
constexpr int CHUNKS_PER_BLOCK = 8;
constexpr int EDGES_PER_BLOCK  = CHUNK * CHUNKS_PER_BLOCK; // 1024

template <bool USE_XT>
__global__ __launch_bounds__(256)
void spmm_scatter_kernel(const long long* __restrict__ idx,   // [2, E] rows then cols (int64)
                         const float*     __restrict__ vals,  // [E]
                         const float*     __restrict__ xsrc,  // xT [N,64] or x [64,N]
                         float*           __restrict__ out,   // [64, N]
                         int E, int N) {
    // Staged as raw bytes; int64 indices are read back as their low dword
    // (all indices < 120000, so the high dword is always zero).
    __shared__ __align__(16) uint32_t rowsL[2][CHUNK * 2];  // 2 x 1024 B
    __shared__ __align__(16) uint32_t colsL[2][CHUNK * 2];  // 2 x 1024 B
    __shared__ __align__(16) float    valsL[2][CHUNK];      // 2 x  512 B

    const int tid = threadIdx.x;          // 0..255
    const int b   = tid & 63;             // batch lane
    const int ty  = tid >> 6;             // edge slot 0..3

    const long long* rowsG = idx;
    const long long* colsG = idx + E;

    const int e0 = blockIdx.x * EDGES_PER_BLOCK;

    const uint32_t maxIdxOff = (uint32_t)E * 8u - 16u;   // clamp keeps reads in-bounds
    const uint32_t maxValOff = (uint32_t)E * 4u - 16u;

    // per-lane hoisted bases
    float*       outb = out + (size_t)(uint32_t)b * (uint32_t)N;  // out[b, :]
    const float* xb   = USE_XT ? (xsrc + b)                       // xT[:, b], stride 64
                               : (xsrc + (size_t)(uint32_t)b * (uint32_t)N); // x[b, :]

    auto stage = [&](int buf, int chunkBase) {
        if (tid < 64) {                                   // rows: 64 lanes x 16 B
            uint32_t off = (uint32_t)chunkBase * 8u + (uint32_t)tid * 16u;
            if (off > maxIdxOff) off = maxIdxOff;
            async_ld16(lds_off_u32(&rowsL[buf][0]) + (uint32_t)tid * 16u, rowsG, off);
        } else if (tid < 128) {                           // cols: 64 lanes x 16 B
            const int t = tid - 64;
            uint32_t off = (uint32_t)chunkBase * 8u + (uint32_t)t * 16u;
            if (off > maxIdxOff) off = maxIdxOff;
            async_ld16(lds_off_u32(&colsL[buf][0]) + (uint32_t)t * 16u, colsG, off);
        } else if (tid < 160) {                           // vals: 32 lanes x 16 B
            const int t = tid - 128;
            uint32_t off = (uint32_t)chunkBase * 4u + (uint32_t)t * 16u;
            if (off > maxValOff) off = maxValOff;
            async_ld16(lds_off_u32(&valsL[buf][0]) + (uint32_t)t * 16u, vals, off);
        }
    };

    stage(0, e0);
    stage(1, e0 + CHUNK);

    for (int k = 0; k < CHUNKS_PER_BLOCK; ++k) {
        // async loads complete in order: <=1 outstanding means chunk k has landed
        if (k == CHUNKS_PER_BLOCK - 1) wait_async_le0(); else wait_async_le1();
        __syncthreads();

        const int buf   = k & 1;
        const int cbase = e0 + k * CHUNK;
        #pragma unroll 8
        for (int s = 0; s < CHUNK / 4; ++s) {
            const int sl = s * 4 + ty;
            const int e  = cbase + sl;
            if (e < E) {
                const uint32_t r = rowsL[buf][2 * sl];   // low dword of int64 (LDS broadcast)
                const uint32_t c = colsL[buf][2 * sl];
                const float    v = valsL[buf][sl];
                float xv;
                if (USE_XT) xv = xb[c * (uint32_t)BATCH];  // coalesced 256 B across wave pair
                else        xv = xb[c];                    // strided fallback (L2-resident)
                atomicAdd(outb + r, v * xv);               // global_atomic_add_f32, no return
            }
        }
        __syncthreads();
        if (k + 2 < CHUNKS_PER_BLOCK) stage(buf, e0 + (k + 2) * CHUNK);
    }
}

// ---------------- host launcher ----------------
extern "C" void kernel_launch(void* const* d_in, const int* in_sizes, int n_in,
                              void* d_out, int out_size, void* d_ws, size_t ws_size,
                              hipStream_t stream) {
    const float*     x    = (const float*)d_in[0];       // [64, N]
    const long long* idx  = (const long long*)d_in[1];   // [2, E] int64
    const float*     vals = (const float*)d_in[2];       // [E]
    float*           out  = (float*)d_out;               // [64, N]

    const int E = in_sizes[2];
    const int N = in_sizes[0] / BATCH;

    // 1) zero the output (graph-capture safe)
    {
        const size_t n4 = (size_t)out_size / 4;
        int blocks = (int)((n4 + 255) / 256);
        if (blocks > 4096) blocks = 4096;
        zero4_kernel<<<blocks, 256, 0, stream>>>((float4*)out, n4);
    }

    // 2) transpose x into workspace if it fits (30.72 MB), else strided fallback
    const size_t xtBytes = (size_t)N * BATCH * sizeof(float);
    const bool useXT = (ws_size >= xtBytes);
    const float* xsrc = x;
    if (useXT) {
        float* xT = (float*)d_ws;
        transpose_x_kernel<<<(N + TN - 1) / TN, 256, 0, stream>>>(x, xT, N);
        xsrc = xT;
    }

    // 3) edge scatter with async-to-LDS double-buffered edge streaming
    const int blocks = (E + EDGES_PER_BLOCK - 1) / EDGES_PER_BLOCK;   // 1875
    if (useXT)
        spmm_scatter_kernel<true ><<<blocks, 256, 0, stream>>>(idx, vals, xsrc, out, E, N);
    else
        spmm_scatter_kernel<false><<<blocks, 256, 0, stream>>>(idx, vals, xsrc, out, E, N);
}